// DepthConv_14620068675753
// MI455X (gfx1250) — compile-verified
//
#include <hip/hip_runtime.h>

// Dynamic depthwise 3x3 conv, padding=1, stride=1. Shapes fixed: N=4, C=64, H=W=256.
// x:  [N, C, H, W] f32
// w:  [N, C*9, H, W] f32   (kernel positions row-major (ki,kj) per channel)
// out[n,c,h,w] = sum_{ki,kj} w[n,c,ki*3+kj,h,w] * xpad[n,c,h-1+ki,w-1+kj]
//
// Memory-bound (AI ~0.41 FLOP/B, 738 MB total => ~32us floor @ 23.3 TB/s).
// Weights (604 MB) streamed once with NT hints; x tile staged via gfx1250
// async global->LDS DMA (ASYNCcnt); compute is VALU FMAs. WMMA inapplicable.

#define TILE_ROWS  4
#define LDS_STRIDE 260   // floats per LDS row: 1040 B, keeps 16B alignment, skews banks

#if defined(__has_builtin)
#  if __has_builtin(__builtin_amdgcn_global_load_async_to_lds_b128)
#    define HAVE_ASYNC_LDS 1
#  endif
#  if __has_builtin(__builtin_amdgcn_s_wait_asynccnt)
#    define HAVE_WAIT_ASYNC 1
#  endif
#endif

typedef float v4f __attribute__((ext_vector_type(4)));
typedef int   v4i __attribute__((ext_vector_type(4)));

// Builtin prototype (confirmed by hipcc diagnostic): (int4* AS1, int4* AS3, imm, imm)
#define GLOBAL_V4I(p) ((__attribute__((address_space(1))) v4i*)(p))
#define LDS_V4I(p)    ((__attribute__((address_space(3))) v4i*)(p))

__global__ __launch_bounds__(256)
void dyn_dwconv3x3(const float* __restrict__ x,
                   const float* __restrict__ wgt,
                   float* __restrict__ out)
{
    constexpr int H = 256, W = 256;
    constexpr int HBLOCKS = H / TILE_ROWS;            // 64 (power of two)
    constexpr size_t PLANE = (size_t)H * W;

    __shared__ float xs[6 * LDS_STRIDE];

    const int plane = blockIdx.x >> 6;                // n*C + c
    const int h0    = (blockIdx.x & (HBLOCKS - 1)) * TILE_ROWS;

    const float* xplane = x   + (size_t)plane * PLANE;
    const float* wbase  = wgt + (size_t)plane * 9 * PLANE;
    float*       oplane = out + (size_t)plane * PLANE;

    const int tid = threadIdx.x;

    // ---- Stage rows h0-1 .. h0+4 of x into LDS (async DMA, b128 chunks) ----
    // 6 rows x 64 chunks = 384 tasks; thread t does t and t+256.
    constexpr int CPR = W / 4;                        // 64 chunks per row
#pragma unroll
    for (int pass = 0; pass < 2; ++pass) {
        const int task = tid + pass * 256;
        if (task < 6 * CPR) {
            const int ry = task >> 6;                 // LDS row 0..5
            const int ck = task & (CPR - 1);
            const int g  = h0 - 1 + ry;               // global x row
            float* lp = &xs[ry * LDS_STRIDE + ck * 4];
            if (g >= 0 && g < H) {
                const float* gp = xplane + (size_t)g * W + ck * 4;
#if HAVE_ASYNC_LDS
                __builtin_amdgcn_global_load_async_to_lds_b128(
                    GLOBAL_V4I(gp), LDS_V4I(lp), /*offset=*/0, /*cpol=*/0);
#else
                *(v4f*)lp = *(const v4f*)gp;
#endif
            } else {
                *(v4f*)lp = (v4f){0.f, 0.f, 0.f, 0.f};  // top/bottom zero padding
            }
        }
    }
#if HAVE_ASYNC_LDS
#  if HAVE_WAIT_ASYNC
    __builtin_amdgcn_s_wait_asynccnt(0);
#  else
    asm volatile("s_wait_asynccnt 0" ::: "memory");
#  endif
#endif
    __syncthreads();

    // ---- Compute: thread (r, cc) -> row h0+r, cols w0..w0+3 ----
    const int r  = tid >> 6;                          // 0..3
    const int cc = tid & 63;                          // 0..63
    const int w0 = cc * 4;
    const int h  = h0 + r;

    const float* wrow = wbase + (size_t)h * W + w0;   // plane k adds k*PLANE

    v4f acc = (v4f){0.f, 0.f, 0.f, 0.f};
#pragma unroll
    for (int ki = 0; ki < 3; ++ki) {
        const float* row = &xs[(r + ki) * LDS_STRIDE];
        // x columns w0-1 .. w0+4 (column padding=1 -> zeros at -1 and W)
        const float vl = (w0 == 0)      ? 0.f : row[w0 - 1];
        const v4f   vm = *(const v4f*)&row[w0];
        const float vr = (w0 + 4 >= W)  ? 0.f : row[w0 + 4];
        const float v[6] = {vl, vm.x, vm.y, vm.z, vm.w, vr};
#pragma unroll
        for (int kj = 0; kj < 3; ++kj) {
            const int k = ki * 3 + kj;
            const v4f* wp = (const v4f*)(wrow + k * PLANE);
            const v4f wv = __builtin_nontemporal_load(wp);   // 604 MB one-shot stream
            acc.x += wv.x * v[0 + kj];
            acc.y += wv.y * v[1 + kj];
            acc.z += wv.z * v[2 + kj];
            acc.w += wv.w * v[3 + kj];
        }
    }
    __builtin_nontemporal_store(acc, (v4f*)(oplane + (size_t)h * W + w0));
}

extern "C" void kernel_launch(void* const* d_in, const int* in_sizes, int n_in,
                              void* d_out, int out_size, void* d_ws, size_t ws_size,
                              hipStream_t stream) {
    const float* x = (const float*)d_in[0];             // [4, 64, 256, 256]
    const float* w = (const float*)d_in[1];             // [4, 64*9, 256, 256]
    float* out = (float*)d_out;                         // [4, 64, 256, 256]

    const int N = 4, C = 64, H = 256;
    (void)in_sizes; (void)n_in; (void)out_size; (void)d_ws; (void)ws_size;

    const int blocks = N * C * (H / TILE_ROWS);         // 16384
    dyn_dwconv3x3<<<blocks, 256, 0, stream>>>(x, w, out);
}